// ProbAttention_23270132810000
// MI455X (gfx1250) — compile-verified
//
#include <hip/hip_runtime.h>

typedef float v2f __attribute__((ext_vector_type(2)));
typedef float v8f __attribute__((ext_vector_type(8)));

#define B_ 4
#define L_ 2048
#define H_ 8
#define D_ 64
#define SK 40
#define U_ 40

// ---------------------------------------------------------------------------
// Kernel 1: M[b,h,i] = max_s(q_i . k_sample_s) - sum_s(...)/L
// ---------------------------------------------------------------------------
__global__ void k_scoreM(const float* __restrict__ Q, const float* __restrict__ K,
                         const int* __restrict__ idx, float* __restrict__ M)
{
    int flat = blockIdx.x * 256 + threadIdx.x;           // bh*L + i
    if (flat >= B_ * H_ * L_) return;
    int i  = flat & (L_ - 1);
    int bh = flat >> 11;
    int b = bh >> 3, h = bh & 7;

    const float4* qp = (const float4*)(Q + ((size_t)(b * L_ + i)) * 512 + h * 64);
    float4 q[16];
    #pragma unroll
    for (int t = 0; t < 16; t++) q[t] = qp[t];

    float mx = -3.0e38f, sm = 0.f;
    for (int s = 0; s < SK; s++) {
        int j = idx[i * SK + s];
        const float4* kp = (const float4*)(K + ((size_t)(b * L_ + j)) * 512 + h * 64);
        float acc = 0.f;
        #pragma unroll
        for (int t = 0; t < 16; t++) {
            float4 kv = kp[t];
            acc += q[t].x * kv.x + q[t].y * kv.y + q[t].z * kv.z + q[t].w * kv.w;
        }
        mx = fmaxf(mx, acc);
        sm += acc;
    }
    M[flat] = mx - sm * (1.0f / (float)L_);
}

// ---------------------------------------------------------------------------
// Kernel 2: top-40 indices per (b,h), lowest-index tie-break (matches top_k)
// ---------------------------------------------------------------------------
__global__ void k_topk(const float* __restrict__ M, int* __restrict__ Mtop)
{
    __shared__ float vals[L_];
    __shared__ float rv[256];
    __shared__ int   ri[256];
    int bh = blockIdx.x, tid = threadIdx.x;
    const float* row = M + (size_t)bh * L_;
    for (int i = tid; i < L_; i += 256) vals[i] = row[i];
    __syncthreads();

    for (int r = 0; r < U_; r++) {
        float bv = -3.0e38f; int bi = 0x7fffffff;
        for (int i = tid; i < L_; i += 256) {
            float v = vals[i];
            if (v > bv || (v == bv && i < bi)) { bv = v; bi = i; }
        }
        rv[tid] = bv; ri[tid] = bi;
        __syncthreads();
        for (int off = 128; off > 0; off >>= 1) {
            if (tid < off) {
                float ov = rv[tid + off]; int oi = ri[tid + off];
                if (ov > rv[tid] || (ov == rv[tid] && oi < ri[tid])) { rv[tid] = ov; ri[tid] = oi; }
            }
            __syncthreads();
        }
        if (tid == 0) { Mtop[bh * U_ + r] = ri[0]; vals[ri[0]] = -3.0e38f; }
        __syncthreads();
    }
}

// ---------------------------------------------------------------------------
// Kernel 3: baseline output = cumsum(V, axis=L), written in (B,L,H,D) layout
// ---------------------------------------------------------------------------
__global__ void k_cumsum(const float* __restrict__ V, float* __restrict__ Out)
{
    __shared__ float tot[4][64];
    int bh = blockIdx.x;
    int b = bh >> 3, h = bh & 7;
    int seg = threadIdx.x >> 6;      // 0..3 segments of 512
    int d   = threadIdx.x & 63;
    const int SEG = L_ / 4;
    size_t base = ((size_t)(b * L_ + seg * SEG)) * 512 + h * 64 + d;

    float acc = 0.f;
    for (int i = 0; i < SEG; i++) {
        acc += V[base + (size_t)i * 512];
        Out[base + (size_t)i * 512] = acc;
    }
    tot[seg][d] = acc;
    __syncthreads();
    float off = 0.f;
    for (int s = 0; s < seg; s++) off += tot[s][d];
    if (seg > 0)
        for (int i = 0; i < SEG; i++)
            Out[base + (size_t)i * 512] += off;
}

// ---------------------------------------------------------------------------
// Kernel 4: flash attention over the 40 selected queries, f32 WMMA 16x16x4.
// 3 waves per block, one 16-row query tile per wave. Causal mask key>qpos.
// V tiles staged via CDNA5 async global->LDS (ASYNCcnt); softmax denominator
// accumulated with an extra WMMA against an all-ones B fragment.
// ---------------------------------------------------------------------------
__global__ void __launch_bounds__(96)
k_attn(const float* __restrict__ Q, const float* __restrict__ K,
       const float* __restrict__ V, const int* __restrict__ Mtop,
       float* __restrict__ Out)
{
    __shared__ float Pbuf[3][16 * 16];
    __shared__ float Vbuf[3][16 * 64];

    int bh = blockIdx.x;
    int b = bh >> 3, h = bh & 7;
    int wave   = threadIdx.x >> 5;       // query tile index 0..2
    int lane   = threadIdx.x & 31;
    int hiHalf = lane >> 4;              // 0 or 1
    int l15    = lane & 15;
    const int* mt = Mtop + bh * U_;

    // A-layout query row for this lane (row = l15 in both half-waves)
    int rowA  = wave * 16 + l15;
    int qrowA = mt[rowA < U_ ? rowA : 0];
    // D-layout causal bound per accumulator register r (row = r + hiHalf*8)
    int qposD[8];
    #pragma unroll
    for (int r = 0; r < 8; r++) {
        int rr = wave * 16 + r + hiHalf * 8;
        qposD[r] = (rr < U_) ? mt[rr] : -1;   // pad rows fully masked
    }

    // Load Q A-fragments: a[kk] covers K-dim cols kk*4 + hiHalf*2 + {0,1}
    v2f qa[16];
    const float* qp = Q + ((size_t)(b * L_ + qrowA)) * 512 + h * 64 + hiHalf * 2;
    #pragma unroll
    for (int kk = 0; kk < 16; kk++) {
        qa[kk].x = qp[kk * 4 + 0];
        qa[kk].y = qp[kk * 4 + 1];
    }

    // Wave-uniform max query position -> early exit over key tiles
    int qmax = qrowA;
    #pragma unroll
    for (int m = 1; m < 32; m <<= 1)
        qmax = max(qmax, __shfl_xor(qmax, m, 32));

    float mx[8];
    v8f o0 = {}, o1 = {}, o2 = {}, o3 = {}, o4 = {};   // o4 = softmax denominator
    #pragma unroll
    for (int r = 0; r < 8; r++) mx[r] = -3.0e38f;

    v2f ones; ones.x = 1.0f; ones.y = 1.0f;

    float* Pb = Pbuf[wave];
    float* Vb = Vbuf[wave];
    // Precompute per-lane LDS/global staging addresses for the V tile:
    // lane covers 8 x 16B chunks: flat float index = (rep*32 + lane)*4
    unsigned vlds[8];
    size_t   vgoff[8];
    #pragma unroll
    for (int rep = 0; rep < 8; rep++) {
        int flat = (rep * 32 + lane) * 4;
        int vr = flat >> 6, vc = flat & 63;
        vlds[rep]  = (unsigned)(size_t)(Vb + flat);                     // LDS offset (low 32b)
        vgoff[rep] = ((size_t)(b * L_ + vr)) * 512 + h * 64 + vc;       // + kb*512 later
    }

    for (int kb = 0; kb <= qmax; kb += 16) {
        // ---- async-stage V tile (16 x 64) into LDS; overlaps the S WMMAs.
        //      s_wait_dscnt first: previous iteration's LDS reads must retire (WAR).
        asm volatile("s_wait_dscnt 0x0" ::: "memory");
        #pragma unroll
        for (int rep = 0; rep < 8; rep++) {
            const float* gp = V + vgoff[rep] + (size_t)kb * 512;
            asm volatile("global_load_async_to_lds_b128 %0, %1, off"
                         :: "v"(vlds[rep]), "v"(gp) : "memory");
        }

        // ---- S = Q * K^T (16 queries x 16 keys), K-dim = 64 in steps of 4
        v8f c = {};
        const float* kp = K + ((size_t)(b * L_ + kb + l15)) * 512 + h * 64 + hiHalf * 2;
        #pragma unroll
        for (int kk = 0; kk < 16; kk++) {
            v2f bf;
            bf.x = kp[kk * 4 + 0];
            bf.y = kp[kk * 4 + 1];
            c = __builtin_amdgcn_wmma_f32_16x16x4_f32(false, qa[kk], false, bf,
                                                      (short)0, c, false, false);
        }

        // ---- mask + online softmax max-update (row r lives in one 16-lane group)
        int kn = kb + l15;
        float p[8];
        #pragma unroll
        for (int r = 0; r < 8; r++) {
            bool maskd = (kn > qposD[r]);
            float s = maskd ? -3.0e38f : c[r] * 0.125f;   // 1/sqrt(64)
            float tmx = s;
            tmx = fmaxf(tmx, __shfl_xor(tmx, 1, 32));
            tmx = fmaxf(tmx, __shfl_xor(tmx, 2, 32));
            tmx = fmaxf(tmx, __shfl_xor(tmx, 4, 32));
            tmx = fmaxf(tmx, __shfl_xor(tmx, 8, 32));
            float nmx = fmaxf(mx[r], tmx);
            float scale = __expf(mx[r] - nmx);
            p[r] = maskd ? 0.f : __expf(s - nmx);
            mx[r] = nmx;
            o0[r] *= scale; o1[r] *= scale; o2[r] *= scale; o3[r] *= scale;
            o4[r] *= scale;                    // denominator rescales identically
        }

        // ---- relayout P (D-layout -> A-layout) through wave-private LDS
        #pragma unroll
        for (int r = 0; r < 8; r++)
            Pb[(r + hiHalf * 8) * 16 + l15] = p[r];
        __builtin_amdgcn_fence(__ATOMIC_ACQ_REL, "wavefront");  // LDS in-order per wave

        v2f pa[4];
        #pragma unroll
        for (int kk2 = 0; kk2 < 4; kk2++) {
            int kc = kk2 * 4 + hiHalf * 2;
            pa[kk2].x = Pb[l15 * 16 + kc + 0];
            pa[kk2].y = Pb[l15 * 16 + kc + 1];
        }

        // ---- wait for async V tile, then O += P(16x16) * V(16x64)
        asm volatile("s_wait_asynccnt 0x0" ::: "memory");
        #pragma unroll
        for (int kk2 = 0; kk2 < 4; kk2++) {
            int kr = kk2 * 4 + hiHalf * 2;
            v2f bv;
            bv.x = Vb[(kr + 0) * 64 +  0 + l15];
            bv.y = Vb[(kr + 1) * 64 +  0 + l15];
            o0 = __builtin_amdgcn_wmma_f32_16x16x4_f32(false, pa[kk2], false, bv, (short)0, o0, false, false);
            bv.x = Vb[(kr + 0) * 64 + 16 + l15];
            bv.y = Vb[(kr + 1) * 64 + 16 + l15];
            o1 = __builtin_amdgcn_wmma_f32_16x16x4_f32(false, pa[kk2], false, bv, (short)0, o1, false, false);
            bv.x = Vb[(kr + 0) * 64 + 32 + l15];
            bv.y = Vb[(kr + 1) * 64 + 32 + l15];
            o2 = __builtin_amdgcn_wmma_f32_16x16x4_f32(false, pa[kk2], false, bv, (short)0, o2, false, false);
            bv.x = Vb[(kr + 0) * 64 + 48 + l15];
            bv.y = Vb[(kr + 1) * 64 + 48 + l15];
            o3 = __builtin_amdgcn_wmma_f32_16x16x4_f32(false, pa[kk2], false, bv, (short)0, o3, false, false);
            // softmax denominator: row-sum of P via all-ones B fragment
            o4 = __builtin_amdgcn_wmma_f32_16x16x4_f32(false, pa[kk2], false, ones, (short)0, o4, false, false);
        }
        __builtin_amdgcn_fence(__ATOMIC_ACQ_REL, "wavefront");  // before LDS reuse next iter
    }

    // ---- normalize and scatter into cumsum baseline (real rows only)
    #pragma unroll
    for (int r = 0; r < 8; r++) {
        int rr = wave * 16 + r + hiHalf * 8;
        if (rr < U_) {
            float inv = 1.0f / o4[r];          // every column of o4 holds the row sum
            float* op = Out + ((size_t)(b * L_ + qposD[r])) * 512 + h * 64;
            op[ 0 + l15] = o0[r] * inv;
            op[16 + l15] = o1[r] * inv;
            op[32 + l15] = o2[r] * inv;
            op[48 + l15] = o3[r] * inv;
        }
    }
}

// ---------------------------------------------------------------------------
extern "C" void kernel_launch(void* const* d_in, const int* in_sizes, int n_in,
                              void* d_out, int out_size, void* d_ws, size_t ws_size,
                              hipStream_t stream)
{
    const float* Q  = (const float*)d_in[0];
    const float* K  = (const float*)d_in[1];
    const float* V  = (const float*)d_in[2];
    const int*  idx = (const int*)d_in[3];
    float* Out = (float*)d_out;

    float* wsM   = (float*)d_ws;                                       // B*H*L floats
    int*   wsTop = (int*)((char*)d_ws + (size_t)B_ * H_ * L_ * sizeof(float)); // B*H*U ints

    k_scoreM<<<(B_ * H_ * L_) / 256, 256, 0, stream>>>(Q, K, idx, wsM);
    k_topk  <<<B_ * H_, 256, 0, stream>>>(wsM, wsTop);
    k_cumsum<<<B_ * H_, 256, 0, stream>>>(V, Out);
    k_attn  <<<B_ * H_,  96, 0, stream>>>(Q, K, V, wsTop, Out);
}